// GraphSAGE_83356725281380
// MI455X (gfx1250) — compile-verified
//
#include <hip/hip_runtime.h>
#include <math.h>

#define N_NODES 50000
#define IN_DIM  128
#define HID     32
#define OUTD    7
#define W1_LDS_STRIDE 130   // 130 % 64 == 2 -> banks differ by 2n across the 16 B-lanes (conflict-free)
#define W2_LDS_STRIDE 34    // 34 % 64 == 34 -> distinct banks across 16 lanes

typedef __attribute__((ext_vector_type(2))) float v2f;
typedef __attribute__((ext_vector_type(8))) float v8f;

static __device__ __forceinline__ v8f wmma_f32_4(v2f a, v2f b, v8f c) {
    // D = A(16x4 f32) * B(4x16 f32) + C(16x16 f32)
    return __builtin_amdgcn_wmma_f32_16x16x4_f32(
        /*neg_a=*/false, a, /*neg_b=*/false, b,
        /*c_mod=*/(short)0, c, /*reuse_a=*/false, /*reuse_b=*/false);
}

// ---------------------------------------------------------------- zero scratch
__global__ void zero_f32(float* __restrict__ p, int n) {
    int i = blockIdx.x * blockDim.x + threadIdx.x;
    if (i < n) p[i] = 0.0f;
}

// ------------------------------------------------ layer-1 linears: x@W1_l^T, x@W1_r^T
// one wave computes a 16-node tile, full 32 output cols, for BOTH weight matrices.
__global__ __launch_bounds__(256) void linear1_wmma(
    const float* __restrict__ x, const float* __restrict__ W1l,
    const float* __restrict__ W1r, float* __restrict__ xl, float* __restrict__ xr)
{
    __shared__ float sW[2][HID][W1_LDS_STRIDE];
    const int tid = threadIdx.x;
    for (int i = tid; i < 2 * HID * IN_DIM; i += 256) {
        int m = i / (HID * IN_DIM);
        int r = (i / IN_DIM) % HID;
        int c = i % IN_DIM;
        sW[m][r][c] = m ? W1r[r * IN_DIM + c] : W1l[r * IN_DIM + c];
    }
    __syncthreads();

    const int wave = (blockIdx.x * 256 + tid) >> 5;
    const int lane = tid & 31;
    if (wave >= N_NODES / 16) return;           // 3125 tiles, uniform per wave

    const int r0  = wave * 16;
    const int hi  = lane >> 4;                  // half of wave
    const int l15 = lane & 15;
    const int khi = hi * 2;                     // K sub-offset for this half

    const float* arow = x + (size_t)(r0 + l15) * IN_DIM + khi;   // A frag source
    const float* bl0  = &sW[0][l15][khi];       // B: cols 0-15 of W1_l^T
    const float* bl1  = &sW[0][l15 + 16][khi];  // B: cols 16-31
    const float* br0  = &sW[1][l15][khi];
    const float* br1  = &sW[1][l15 + 16][khi];

    v8f aL0 = {}, aL1 = {}, aR0 = {}, aR1 = {};
#pragma unroll
    for (int k = 0; k < IN_DIM; k += 4) {
        v2f a  = *(const v2f*)(arow + k);
        v2f b0 = *(const v2f*)(bl0 + k);
        v2f b1 = *(const v2f*)(bl1 + k);
        v2f b2 = *(const v2f*)(br0 + k);
        v2f b3 = *(const v2f*)(br1 + k);
        aL0 = wmma_f32_4(a, b0, aL0);
        aL1 = wmma_f32_4(a, b1, aL1);
        aR0 = wmma_f32_4(a, b2, aR0);
        aR1 = wmma_f32_4(a, b3, aR1);
    }

    const int col   = l15;
    const int mbase = r0 + hi * 8;
#pragma unroll
    for (int r = 0; r < 8; ++r) {
        int node = mbase + r;
        xl[node * HID + col]      = aL0[r];
        xl[node * HID + 16 + col] = aL1[r];
        xr[node * HID + col]      = aR0[r];
        xr[node * HID + 16 + col] = aR1[r];
    }
}

// ---------------------------------------------- layer-1 edge aggregation (32-dim)
// 8 threads per edge, each handles a float4 chunk; per-edge degree from lane 0.
__global__ void edge_agg1(const int* __restrict__ src, const int* __restrict__ dst,
                          const float* __restrict__ xl, float* __restrict__ agg1,
                          float* __restrict__ deg, int E)
{
    long long t = (long long)blockIdx.x * blockDim.x + threadIdx.x;
    int e = (int)(t >> 3);
    if (e >= E) return;
    int c = ((int)t & 7) * 4;
    int s = src[e], d = dst[e];
    float4 v = *(const float4*)(xl + (size_t)s * HID + c);
    atomicAdd(&agg1[(size_t)d * HID + c + 0], v.x);
    atomicAdd(&agg1[(size_t)d * HID + c + 1], v.y);
    atomicAdd(&agg1[(size_t)d * HID + c + 2], v.z);
    atomicAdd(&agg1[(size_t)d * HID + c + 3], v.w);
    if (((int)t & 7) == 0) atomicAdd(&deg[d], 1.0f);
}

// ------------------------- combine, L2-normalize, emit embedding + relu(h)
__global__ void combine1(const float* __restrict__ agg1, const float* __restrict__ deg,
                         const float* __restrict__ xr, const float* __restrict__ b1,
                         float* __restrict__ emb, float* __restrict__ rh)
{
    int t = blockIdx.x * blockDim.x + threadIdx.x;
    int node = t >> 5;
    int j = t & 31;
    if (node >= N_NODES) return;
    float inv = 1.0f / fmaxf(deg[node], 1.0f);
    float v = agg1[(size_t)node * HID + j] * inv + b1[j] + xr[(size_t)node * HID + j];
    float sq = v * v;
#pragma unroll
    for (int o = 16; o > 0; o >>= 1) sq += __shfl_xor(sq, o, 32);
    float nrm = sqrtf(sq);
    float h = v / fmaxf(nrm, 1e-12f);
    emb[(size_t)node * HID + j] = h;
    rh[(size_t)node * HID + j]  = fmaxf(h, 0.0f);
}

// ------------------------------------ layer-2 linears: rh@W2_l^T, rh@W2_r^T (7 cols, padded to 16)
__global__ __launch_bounds__(256) void linear2_wmma(
    const float* __restrict__ rh, const float* __restrict__ W2l,
    const float* __restrict__ W2r, float* __restrict__ hl, float* __restrict__ hr)
{
    __shared__ float sW[2][16][W2_LDS_STRIDE];
    const int tid = threadIdx.x;
    for (int i = tid; i < 2 * 16 * W2_LDS_STRIDE; i += 256) ((float*)sW)[i] = 0.0f;
    __syncthreads();
    for (int i = tid; i < 2 * OUTD * HID; i += 256) {
        int m = i / (OUTD * HID);
        int r = (i / HID) % OUTD;
        int c = i % HID;
        sW[m][r][c] = m ? W2r[r * HID + c] : W2l[r * HID + c];
    }
    __syncthreads();

    const int wave = (blockIdx.x * 256 + tid) >> 5;
    const int lane = tid & 31;
    if (wave >= N_NODES / 16) return;

    const int r0  = wave * 16;
    const int hi  = lane >> 4;
    const int l15 = lane & 15;
    const int khi = hi * 2;

    const float* arow = rh + (size_t)(r0 + l15) * HID + khi;
    const float* bl   = &sW[0][l15][khi];
    const float* br   = &sW[1][l15][khi];

    v8f aL = {}, aR = {};
#pragma unroll
    for (int k = 0; k < HID; k += 4) {
        v2f a  = *(const v2f*)(arow + k);
        v2f b0 = *(const v2f*)(bl + k);
        v2f b1 = *(const v2f*)(br + k);
        aL = wmma_f32_4(a, b0, aL);
        aR = wmma_f32_4(a, b1, aR);
    }

    const int col   = l15;
    const int mbase = r0 + hi * 8;
#pragma unroll
    for (int r = 0; r < 8; ++r) {
        hl[(size_t)(mbase + r) * 16 + col] = aL[r];
        hr[(size_t)(mbase + r) * 16 + col] = aR[r];
    }
}

// ---------------------------------------------- layer-2 edge aggregation (7-dim)
__global__ void edge_agg2(const int* __restrict__ src, const int* __restrict__ dst,
                          const float* __restrict__ hl, float* __restrict__ agg2, int E)
{
    long long t = (long long)blockIdx.x * blockDim.x + threadIdx.x;
    int e = (int)(t >> 3);
    if (e >= E) return;
    int j = (int)t & 7;
    if (j >= OUTD) return;
    int s = src[e], d = dst[e];
    atomicAdd(&agg2[(size_t)d * 8 + j], hl[(size_t)s * 16 + j]);
}

// -------------------------------------------------- combine + log_softmax
__global__ void final_out(const float* __restrict__ agg2, const float* __restrict__ deg,
                          const float* __restrict__ hr, const float* __restrict__ b2,
                          float* __restrict__ out)
{
    int n = blockIdx.x * blockDim.x + threadIdx.x;
    if (n >= N_NODES) return;
    float inv = 1.0f / fmaxf(deg[n], 1.0f);
    float o[OUTD];
    float m = -INFINITY;
#pragma unroll
    for (int j = 0; j < OUTD; ++j) {
        o[j] = agg2[(size_t)n * 8 + j] * inv + b2[j] + hr[(size_t)n * 16 + j];
        m = fmaxf(m, o[j]);
    }
    float s = 0.0f;
#pragma unroll
    for (int j = 0; j < OUTD; ++j) s += expf(o[j] - m);
    float lse = m + logf(s);
#pragma unroll
    for (int j = 0; j < OUTD; ++j) out[(size_t)n * OUTD + j] = o[j] - lse;
}

extern "C" void kernel_launch(void* const* d_in, const int* in_sizes, int n_in,
                              void* d_out, int out_size, void* d_ws, size_t ws_size,
                              hipStream_t stream)
{
    const float* x    = (const float*)d_in[0];
    const int*   esrc = (const int*)d_in[1];
    const int*   edst = (const int*)d_in[2];
    const float* W1l  = (const float*)d_in[3];
    const float* b1   = (const float*)d_in[4];
    const float* W1r  = (const float*)d_in[5];
    const float* W2l  = (const float*)d_in[6];
    const float* b2   = (const float*)d_in[7];
    const float* W2r  = (const float*)d_in[8];
    const int E = in_sizes[1];

    // workspace layout (floats)
    float* ws   = (float*)d_ws;
    float* agg1 = ws;                          // N*32   (zeroed)
    float* deg  = agg1 + (size_t)N_NODES * HID; // N     (zeroed)
    float* agg2 = deg + N_NODES;               // N*8    (zeroed)
    float* xl   = agg2 + (size_t)N_NODES * 8;  // N*32
    float* xr   = xl + (size_t)N_NODES * HID;  // N*32
    float* rh   = xr + (size_t)N_NODES * HID;  // N*32
    float* hl   = rh + (size_t)N_NODES * HID;  // N*16
    float* hr   = hl + (size_t)N_NODES * 16;   // N*16

    float* emb  = (float*)d_out;               // N*32
    float* outp = emb + (size_t)N_NODES * HID; // N*7

    const int zero_n = N_NODES * (HID + 1 + 8);
    zero_f32<<<(zero_n + 255) / 256, 256, 0, stream>>>(ws, zero_n);

    const int tiles  = N_NODES / 16;           // 3125
    const int blocks = (tiles + 7) / 8;        // 8 waves/block

    linear1_wmma<<<blocks, 256, 0, stream>>>(x, W1l, W1r, xl, xr);
    {
        long long th = (long long)E * 8;
        edge_agg1<<<(unsigned)((th + 255) / 256), 256, 0, stream>>>(esrc, edst, xl, agg1, deg, E);
    }
    combine1<<<(N_NODES * 32 + 255) / 256, 256, 0, stream>>>(agg1, deg, xr, b1, emb, rh);
    linear2_wmma<<<blocks, 256, 0, stream>>>(rh, W2l, W2r, hl, hr);
    {
        long long th = (long long)E * 8;
        edge_agg2<<<(unsigned)((th + 255) / 256), 256, 0, stream>>>(esrc, edst, hl, agg2, E);
    }
    final_out<<<(N_NODES + 255) / 256, 256, 0, stream>>>(agg2, deg, hr, b2, outp);
}